// NoisyTopkRouter_29506425324173
// MI455X (gfx1250) — compile-verified
//
#include <hip/hip_runtime.h>

// ---------------------------------------------------------------------------
// MoE router for MI455X (gfx1250, wave32):
//   logits = x @ W^T + b  (x: [32768,1024] f32, W: [64,1024] f32, b: [64])
//   top-8 per token, sparse softmax -> probs [32768,64] + indices [32768,8]
//
//   * W staged once per workgroup into LDS, row-major with per-row column
//     rotation (k+4n)&1023 -> bank-conflict-free b64 B-fragment reads with
//     ZERO padding (262,144 B).
//   * Each wave: 32 tokens x 64 experts via V_WMMA_F32_16X16X4_F32.
//     Per k-step: 2 A global b64 + 4 shared B ds b64 + 8 WMMAs.
//   * C tiles -> rotated per-wave LDS scratch -> all 32 lanes stream their
//     token's 64 logits back as b64 pairs and run a branchless sorted
//     insertion top-8 (monotonic-uint keys, lowest-index tie-break),
//     sparse softmax, scatter probs + indices.
//   * LDS = 262,144 + 65,536 = 327,680 B (= 320 KB WGP maximum).
// ---------------------------------------------------------------------------

#define DIMK 1024
#define NEXP 64
#define TOPK 8
#define WAVES_PER_BLOCK 8
#define TOKENS_PER_WAVE 32
#define TOKENS_PER_BLOCK (WAVES_PER_BLOCK * TOKENS_PER_WAVE)  // 256
#define NTOK (8 * 4096)                                       // 32768 tokens

typedef __attribute__((ext_vector_type(2))) float v2f;
typedef __attribute__((ext_vector_type(8))) float v8f;

__device__ __forceinline__ v8f wmma_f32_16x16x4(v2f a, v2f b, v8f c) {
    // 8 args: (neg_a, A, neg_b, B, c_mod, C, reuse_a, reuse_b)
    return __builtin_amdgcn_wmma_f32_16x16x4_f32(false, a, false, b, (short)0, c,
                                                 false, false);
}

// Monotonic key: larger float <=> larger uint (2 VALU ops).
__device__ __forceinline__ unsigned mono_key(float f) {
    const unsigned u = __float_as_uint(f);
    return u ^ ((unsigned)((int)u >> 31) | 0x80000000u);
}

__global__ __launch_bounds__(256) void
router_topk_kernel(const float* __restrict__ X, const float* __restrict__ W,
                   const float* __restrict__ Bv, float* __restrict__ outP,
                   float* __restrict__ outI) {
    extern __shared__ float lds[];
    float* ldsW = lds;                 // [64][1024], per-row rotated by 4n floats
    float* ldsL = lds + NEXP * DIMK;   // [8 waves][32][64], rotated by 2m

    const int tid   = threadIdx.x;
    const int wave  = tid >> 5;
    const int lane  = tid & 31;
    const int halfk = lane >> 4;   // 0: K=k0..k0+1, 1: K=k0+2..k0+3
    const int lrow  = lane & 15;   // M index (A) / N index (B,C)

    // ---- Stage W [64 x 1024] into LDS, rotating each row by 4n floats -----
    for (int c = tid; c < (NEXP * DIMK / 4); c += 256) {
        const int n  = c >> 8;           // 256 float4 chunks per row
        const int k4 = (c & 255) * 4;
        const float4 v = ((const float4*)W)[c];
        const int col = (k4 + 4 * n) & (DIMK - 1);  // multiple of 4, no wrap
        float* dst = ldsW + n * DIMK + col;
        dst[0] = v.x; dst[1] = v.y; dst[2] = v.z; dst[3] = v.w;
    }
    __syncthreads();

    // ---- WMMA GEMM: 32 tokens x 64 experts per wave -----------------------
    const int tileBase = (blockIdx.x * WAVES_PER_BLOCK + wave) * TOKENS_PER_WAVE;

    const float* arow0 = X + (size_t)(tileBase + lrow) * DIMK + 2 * halfk;
    const float* arow1 = arow0 + (size_t)16 * DIMK;

    int nrow[4], brot[4];
#pragma unroll
    for (int nt = 0; nt < 4; ++nt) {
        nrow[nt] = nt * 16 + lrow;
        brot[nt] = 2 * halfk + 4 * nrow[nt];
    }

    v8f acc[2][4];
#pragma unroll
    for (int t = 0; t < 2; ++t)
#pragma unroll
        for (int nt = 0; nt < 4; ++nt) acc[t][nt] = (v8f){};

#pragma unroll 4
    for (int k0 = 0; k0 < DIMK; k0 += 4) {
        const v2f a0 = *(const v2f*)(arow0 + k0);
        const v2f a1 = *(const v2f*)(arow1 + k0);
        v2f w[4];
#pragma unroll
        for (int nt = 0; nt < 4; ++nt) {
            const int col = (k0 + brot[nt]) & (DIMK - 1);  // even -> 8B aligned
            w[nt] = *(const v2f*)(ldsW + nrow[nt] * DIMK + col);
        }
#pragma unroll
        for (int nt = 0; nt < 4; ++nt) {
            acc[0][nt] = wmma_f32_16x16x4(a0, w[nt], acc[0][nt]);
            acc[1][nt] = wmma_f32_16x16x4(a1, w[nt], acc[1][nt]);
        }
    }

    // ---- Bias add (N = lane%16 for every C VGPR) --------------------------
#pragma unroll
    for (int nt = 0; nt < 4; ++nt) {
        const float bb = Bv[nt * 16 + lrow];
#pragma unroll
        for (int v = 0; v < 8; ++v) { acc[0][nt][v] += bb; acc[1][nt][v] += bb; }
    }

    // ---- Spill C tiles to rotated per-wave logits scratch -----------------
    // C layout: VGPR v, lanes 0-15: M=v, N=lane; lanes 16-31: M=v+8, N=lane-16
    float* Lw = ldsL + wave * TOKENS_PER_WAVE * NEXP;
#pragma unroll
    for (int t = 0; t < 2; ++t)
#pragma unroll
        for (int nt = 0; nt < 4; ++nt)
#pragma unroll
            for (int v = 0; v < 8; ++v) {
                const int m = t * 16 + v + 8 * halfk;           // token in wave
                const int col = (nrow[nt] + 2 * m) & (NEXP - 1);
                Lw[(m << 6) + col] = acc[t][nt][v];
            }

    // Wave-private scratch: wave-level DS ordering before cross-lane reads.
    asm volatile("s_wait_dscnt 0" ::: "memory");

    // ---- Top-8 + sparse softmax: all 32 lanes, one token each -------------
    {
        const int m     = lane;                  // token within wave
        const int token = tileBase + m;
        const float* Lr = Lw + (m << 6);
        const int rot   = 2 * m;

        // Sorted 8-entry register heap (descending). Key 0 < any real logit.
        unsigned tK[TOPK];
        int      tI[TOPK];
#pragma unroll
        for (int j = 0; j < TOPK; ++j) { tK[j] = 0u; tI[j] = 0; }

        // Stream the 64 logits as rotated b64 pairs (conflict-free: lane m
        // hits banks {e+2m, e+2m+1}; e,2m even -> 8-byte aligned, no wrap).
#pragma unroll
        for (int e = 0; e < NEXP; e += 2) {
            const v2f pr = *(const v2f*)(Lr + ((e + rot) & (NEXP - 1)));
#pragma unroll
            for (int sub = 0; sub < 2; ++sub) {
                const unsigned key = mono_key(pr[sub]);
                const int      idx = e + sub;
                // branchless sorted insertion; strict '>' keeps the earlier
                // (lower) index ahead on ties, matching jax.lax.top_k.
                bool c[TOPK];
#pragma unroll
                for (int j = 0; j < TOPK; ++j) c[j] = key > tK[j];
#pragma unroll
                for (int j = TOPK - 1; j >= 1; --j) {
                    tK[j] = c[j] ? (c[j - 1] ? tK[j - 1] : key) : tK[j];
                    tI[j] = c[j] ? (c[j - 1] ? tI[j - 1] : idx) : tI[j];
                }
                tK[0] = c[0] ? key : tK[0];
                tI[0] = c[0] ? idx : tI[0];
            }
        }

        // Recover exact f32 logits from keys; softmax over the 8 kept
        // (everything else is exp(-inf) = 0).
        float topv[TOPK];
#pragma unroll
        for (int j = 0; j < TOPK; ++j) {
            const unsigned bk = tK[j];
            const unsigned orig = (bk & 0x80000000u) ? (bk ^ 0x80000000u) : ~bk;
            topv[j] = __uint_as_float(orig);
        }
        const float mx = topv[0];
        float p[TOPK];
        float s = 0.0f;
#pragma unroll
        for (int j = 0; j < TOPK; ++j) {
            p[j] = __expf(topv[j] - mx);
            s += p[j];
        }
        const float inv = 1.0f / s;

        // zero-fill the 64-wide prob row, then scatter the 8 probabilities
        float* orow = outP + (size_t)token * NEXP;
        const float4 z = {0.0f, 0.0f, 0.0f, 0.0f};
#pragma unroll
        for (int q = 0; q < NEXP / 4; ++q) ((float4*)orow)[q] = z;
#pragma unroll
        for (int j = 0; j < TOPK; ++j) orow[tI[j]] = p[j] * inv;

        float* irow = outI + (size_t)token * TOPK;
#pragma unroll
        for (int j = 0; j < TOPK; ++j) irow[j] = (float)tI[j];
    }
}

extern "C" void kernel_launch(void* const* d_in, const int* in_sizes, int n_in,
                              void* d_out, int out_size, void* d_ws, size_t ws_size,
                              hipStream_t stream) {
    (void)in_sizes; (void)n_in; (void)d_ws; (void)ws_size; (void)out_size;
    const float* X  = (const float*)d_in[0];
    const float* W  = (const float*)d_in[1];
    const float* Bv = (const float*)d_in[2];

    float* outP = (float*)d_out;                        // [32768*64] probs
    float* outI = (float*)d_out + (size_t)NTOK * NEXP;  // [32768*8] indices (as float)

    const size_t shmem =
        (size_t)(NEXP * DIMK + WAVES_PER_BLOCK * TOKENS_PER_WAVE * NEXP) *
        sizeof(float);  // 327,680 B == 320 KB WGP LDS maximum

    static_assert(NTOK % TOKENS_PER_BLOCK == 0, "grid tiling");
    hipFuncSetAttribute((const void*)router_topk_kernel,
                        hipFuncAttributeMaxDynamicSharedMemorySize, (int)shmem);
    router_topk_kernel<<<NTOK / TOKENS_PER_BLOCK, 256, shmem, stream>>>(X, W, Bv,
                                                                       outP, outI);
}